// LengthRegulator_8169027797081
// MI455X (gfx1250) — compile-verified
//
#include <hip/hip_runtime.h>
#include <stdint.h>

// LengthRegulator for MI455X (gfx1250).
// B=32 rows, T_TXT=512 tokens/row. One workgroup (512 threads = 16 wave32) per row.
// Latency-bound problem: fuse round -> mask -> cumsum -> expand into one kernel.
// Uses CDNA5 async global->LDS copy (ASYNCcnt) + wave32 shuffle scan.

#define T_TXT 512
#define NWAVES 16  // 512 / 32

__global__ __launch_bounds__(T_TXT)
void length_regulator_kernel(const float* __restrict__ dur,
                             const int*   __restrict__ dur_padding,
                             float*       __restrict__ out,
                             int T_mel)
{
    __shared__ float s_dur[T_TXT];
    __shared__ int   s_wave[NWAVES];

    const int tid  = threadIdx.x;
    const int b    = blockIdx.x;
    const int lane = tid & 31;
    const int wid  = tid >> 5;

    // ---- CDNA5 async copy: global -> LDS, tracked by ASYNCcnt ----
    {
        const unsigned long long gaddr =
            (unsigned long long)(uintptr_t)(dur + (size_t)b * T_TXT + tid);
        const unsigned lds_addr = (unsigned)(uintptr_t)&s_dur[tid];
        asm volatile("global_load_async_to_lds_b32 %0, %1, off"
                     :: "v"(lds_addr), "v"(gaddr)
                     : "memory");
    }

    // Overlap: fetch padding through the normal load path while async is in flight.
    const int pad = dur_padding[(size_t)b * T_TXT + tid];

    asm volatile("s_wait_asynccnt 0" ::: "memory");
    __syncthreads();

    // round-half-even (matches jnp.round); inputs are integral so exact anyway.
    int d = (int)__builtin_rintf(s_dur[tid]);
    d *= (1 - pad);

    // ---- inclusive prefix sum: wave32 shuffle scan ----
    int inc = d;
#pragma unroll
    for (int off = 1; off < 32; off <<= 1) {
        int n = __shfl_up(inc, off, 32);
        if (lane >= off) inc += n;
    }
    if (lane == 31) s_wave[wid] = inc;   // per-wave totals
    __syncthreads();

    // scan the 16 wave totals with the first wave
    if (wid == 0) {
        int w = (lane < NWAVES) ? s_wave[lane] : 0;
#pragma unroll
        for (int off = 1; off < NWAVES; off <<= 1) {
            int n = __shfl_up(w, off, 32);
            if (lane >= off) w += n;
        }
        if (lane < NWAVES) s_wave[lane] = w;
    }
    __syncthreads();

    const int cum   = inc + (wid ? s_wave[wid - 1] : 0);  // inclusive cumsum
    const int prev  = cum - d;                            // exclusive cumsum
    const int total = s_wave[NWAVES - 1];                 // row total duration

    float* orow = out + (size_t)b * T_mel;
    const float tok = (float)(tid + 1);                   // 1-based token index

    // Expand: frames [prev, cum) get this token's index (<= DUR_MAX-1 = 7 stores).
    for (int t = prev; t < cum; ++t) orow[t] = tok;

    // Frames past this row's total are 0 (disjoint from the scatter range).
    for (int t = total + tid; t < T_mel; t += T_TXT) orow[t] = 0.0f;
}

extern "C" void kernel_launch(void* const* d_in, const int* in_sizes, int n_in,
                              void* d_out, int out_size, void* d_ws, size_t ws_size,
                              hipStream_t stream) {
    const float* dur = (const float*)d_in[0];
    const int*   pad = (const int*)d_in[1];
    float*       out = (float*)d_out;

    const int B     = in_sizes[0] / T_TXT;  // 32
    const int T_mel = out_size / B;

    length_regulator_kernel<<<B, T_TXT, 0, stream>>>(dur, pad, out, T_mel);
}